// MPGATConv_16484084483542
// MI455X (gfx1250) — compile-verified
//
#include <hip/hip_runtime.h>

typedef float v2f __attribute__((ext_vector_type(2)));
typedef float v8f __attribute__((ext_vector_type(8)));

#define IN_F 128
#define OUT_C 128          // HEADS * OUT_FEATS
#define HEADS 4
#define OUT_F 32
#define NEG_SLOPE 0.2f
#define TILE_M 64
#define APAD 4
#define BPAD 4

// ---------------- init: out = bias broadcast, denom = 0, emax = 0 ----------------
__global__ __launch_bounds__(256) void gat_init(float* __restrict__ out,
                                                float* __restrict__ denom,
                                                unsigned* __restrict__ emax,
                                                const float* __restrict__ bias,
                                                int n_nodes) {
  int i = blockIdx.x * blockDim.x + threadIdx.x;
  if (i < n_nodes * OUT_C) out[i] = bias[i & (OUT_C - 1)];
  if (i < n_nodes * HEADS) {
    denom[i] = 0.0f;
    emax[i] = 0u;  // order-encoding of "most negative"
  }
}

// ---------------- fused GEMM (fp32 WMMA) + el/er ----------------
__global__ __launch_bounds__(128) void gat_gemm(
    const float* __restrict__ feat, const float* __restrict__ W,
    const float* __restrict__ attn_l, const float* __restrict__ attn_r,
    float* __restrict__ feat_src, float* __restrict__ el, float* __restrict__ er,
    int n_nodes) {
  __shared__ float Alds[TILE_M][IN_F + APAD];   // feat tile, later reused for output tile
  __shared__ float Blds[IN_F][16 + BPAD];       // 16-column chunk of W

  const int tid = threadIdx.x;
  const int lane = tid & 31;
  const int wave = tid >> 5;                    // 0..3, each wave owns 16 rows
  const int rowBase = blockIdx.x * TILE_M;

  // Stage feat tile (coalesced), zero-pad past end of node array.
  for (int i = tid; i < TILE_M * IN_F; i += 128) {
    int r = i >> 7, c = i & 127;
    int g = rowBase + r;
    Alds[r][c] = (g < n_nodes) ? feat[(size_t)g * IN_F + c] : 0.0f;
  }

  v8f zero = {};
  v8f acc[8];
#pragma unroll
  for (int t = 0; t < 8; ++t) acc[t] = zero;

  // WMMA f32 16x16x4 lane mapping (ISA 7.12.2):
  //   A (16x4): lanes 0-15 -> M=lane, v[0]=K0 v[1]=K1 ; lanes 16-31 -> K2/K3
  //   B (4x16): lanes 0-15 -> N=lane, v[0]=K0 v[1]=K1 ; lanes 16-31 -> K2/K3
  const int mrow = wave * 16 + (lane & 15);
  const int kk   = (lane >> 4) * 2;
  const int ncol = lane & 15;

#pragma unroll
  for (int t = 0; t < 8; ++t) {
    __syncthreads();
    // Stream W columns [t*16, t*16+16) into LDS.
    for (int i = tid; i < IN_F * 16; i += 128) {
      int r = i >> 4, c = i & 15;
      Blds[r][c] = W[(size_t)r * OUT_C + t * 16 + c];
    }
    __syncthreads();
#pragma unroll 4
    for (int k0 = 0; k0 < IN_F; k0 += 4) {
      v2f a, b;
      a.x = Alds[mrow][k0 + kk];
      a.y = Alds[mrow][k0 + kk + 1];
      b.x = Blds[k0 + kk][ncol];
      b.y = Blds[k0 + kk + 1][ncol];
      acc[t] = __builtin_amdgcn_wmma_f32_16x16x4_f32(
          false, a, false, b, (short)0, acc[t], false, false);
    }
  }

  // Write accumulators back through LDS (C/D layout: VGPR r -> M=r (+8 for hi lanes)).
  __syncthreads();
  const int mw = wave * 16 + (lane >> 4) * 8;
#pragma unroll
  for (int t = 0; t < 8; ++t)
#pragma unroll
    for (int r = 0; r < 8; ++r)
      Alds[mw + r][t * 16 + ncol] = acc[t][r];
  __syncthreads();

  // Coalesced global store of feat_src tile.
  for (int i = tid; i < TILE_M * OUT_C; i += 128) {
    int r = i >> 7, c = i & 127;
    int g = rowBase + r;
    if (g < n_nodes) feat_src[(size_t)g * OUT_C + c] = Alds[r][c];
  }
  // Fused el/er: per (node, head) 32-float dot products.
  for (int p = tid; p < TILE_M * HEADS; p += 128) {
    int r = p >> 2, h = p & 3;
    int g = rowBase + r;
    if (g < n_nodes) {
      float sl = 0.f, sr = 0.f;
#pragma unroll
      for (int f = 0; f < OUT_F; ++f) {
        float v = Alds[r][h * OUT_F + f];
        sl += v * attn_l[h * OUT_F + f];
        sr += v * attn_r[h * OUT_F + f];
      }
      el[(size_t)g * HEADS + h] = sl;
      er[(size_t)g * HEADS + h] = sr;
    }
  }
}

// ---------------- order-preserving fp32 <-> u32 for atomic segment-max ----------------
__device__ __forceinline__ unsigned enc_ord(float x) {
  unsigned b = __float_as_uint(x);
  return (b & 0x80000000u) ? ~b : (b | 0x80000000u);
}
__device__ __forceinline__ float dec_ord(unsigned u) {
  return (u & 0x80000000u) ? __uint_as_float(u & 0x7fffffffu)
                           : __uint_as_float(~u);
}

// ---------------- edge logits + leaky relu + segment max ----------------
__global__ __launch_bounds__(256) void gat_edge_logits(
    const float* __restrict__ el, const float* __restrict__ er,
    const int* __restrict__ esrc, const int* __restrict__ edst,
    float* __restrict__ ebuf, unsigned* __restrict__ emax, int n_edges) {
  int e = blockIdx.x * blockDim.x + threadIdx.x;
  if (e >= n_edges) return;
  int s = esrc[e], d = edst[e];
  const float4 l = *(const float4*)(el + (size_t)s * HEADS);
  const float4 r = *(const float4*)(er + (size_t)d * HEADS);
  float4 v;
  v.x = l.x + r.x; v.y = l.y + r.y; v.z = l.z + r.z; v.w = l.w + r.w;
  v.x = v.x > 0.f ? v.x : NEG_SLOPE * v.x;
  v.y = v.y > 0.f ? v.y : NEG_SLOPE * v.y;
  v.z = v.z > 0.f ? v.z : NEG_SLOPE * v.z;
  v.w = v.w > 0.f ? v.w : NEG_SLOPE * v.w;
  *(float4*)(ebuf + (size_t)e * HEADS) = v;
  unsigned* m = emax + (size_t)d * HEADS;
  atomicMax(m + 0, enc_ord(v.x));
  atomicMax(m + 1, enc_ord(v.y));
  atomicMax(m + 2, enc_ord(v.z));
  atomicMax(m + 3, enc_ord(v.w));
}

// ---------------- exp(e - max) + segment sum of denominators ----------------
__global__ __launch_bounds__(256) void gat_edge_exp(
    const int* __restrict__ edst, const unsigned* __restrict__ emax,
    float* __restrict__ ebuf, float* __restrict__ denom, int n_edges) {
  int e = blockIdx.x * blockDim.x + threadIdx.x;
  if (e >= n_edges) return;
  int d = edst[e];
  float4 v = *(const float4*)(ebuf + (size_t)e * HEADS);
  const unsigned* m = emax + (size_t)d * HEADS;
  v.x = expf(v.x - dec_ord(m[0]));
  v.y = expf(v.y - dec_ord(m[1]));
  v.z = expf(v.z - dec_ord(m[2]));
  v.w = expf(v.w - dec_ord(m[3]));
  *(float4*)(ebuf + (size_t)e * HEADS) = v;
  float* dn = denom + (size_t)d * HEADS;
  unsafeAtomicAdd(dn + 0, v.x);
  unsafeAtomicAdd(dn + 1, v.y);
  unsafeAtomicAdd(dn + 2, v.z);
  unsafeAtomicAdd(dn + 3, v.w);
}

// ---------------- aggregation: one wave32 per edge, lane owns 4 contiguous feats ----------------
__global__ __launch_bounds__(256) void gat_aggregate(
    const float* __restrict__ feat_src, const float* __restrict__ ebuf,
    const float* __restrict__ denom,
    const int* __restrict__ esrc, const int* __restrict__ edst,
    float* __restrict__ out, int n_edges) {
  int gt = blockIdx.x * blockDim.x + threadIdx.x;
  int e = gt >> 5;
  int lane = gt & 31;
  if (e >= n_edges) return;
  int s = esrc[e], d = edst[e];
  int h = lane >> 3;  // lane*4 spans [h*32, h*32+31]
  float a = ebuf[(size_t)e * HEADS + h] /
            fmaxf(denom[(size_t)d * HEADS + h], 1e-38f);
  const float4 f = *(const float4*)(feat_src + (size_t)s * OUT_C + lane * 4);
  float* o = out + (size_t)d * OUT_C + lane * 4;
  unsafeAtomicAdd(o + 0, f.x * a);
  unsafeAtomicAdd(o + 1, f.y * a);
  unsafeAtomicAdd(o + 2, f.z * a);
  unsafeAtomicAdd(o + 3, f.w * a);
}

extern "C" void kernel_launch(void* const* d_in, const int* in_sizes, int n_in,
                              void* d_out, int out_size, void* d_ws, size_t ws_size,
                              hipStream_t stream) {
  const float* feat   = (const float*)d_in[0];
  const float* W      = (const float*)d_in[1];
  const float* attn_l = (const float*)d_in[2];
  const float* attn_r = (const float*)d_in[3];
  const float* bias   = (const float*)d_in[4];
  const int*   esrc   = (const int*)d_in[5];
  const int*   edst   = (const int*)d_in[6];

  const int N = in_sizes[0] / IN_F;
  const int E = in_sizes[5];
  float* out = (float*)d_out;

  // Workspace carve-up (all offsets 16B-aligned).
  float* ws = (float*)d_ws;
  size_t o = 0;
  float* feat_src = ws + o;      o += (size_t)N * OUT_C;   // N*128 fp32
  float* el       = ws + o;      o += (size_t)N * HEADS;
  float* er       = ws + o;      o += (size_t)N * HEADS;
  float* ebuf     = ws + o;      o += (size_t)E * HEADS;
  float* denom    = ws + o;      o += (size_t)N * HEADS;
  unsigned* emax  = (unsigned*)(ws + o);

  // 1) init outputs / reductions
  {
    int total = N * OUT_C;
    gat_init<<<(total + 255) / 256, 256, 0, stream>>>(out, denom, emax, bias, N);
  }
  // 2) WMMA GEMM + el/er
  gat_gemm<<<(N + TILE_M - 1) / TILE_M, 128, 0, stream>>>(
      feat, W, attn_l, attn_r, feat_src, el, er, N);
  // 3) edge logits + segment max
  gat_edge_logits<<<(E + 255) / 256, 256, 0, stream>>>(
      el, er, esrc, edst, ebuf, emax, E);
  // 4) exp + denominators
  gat_edge_exp<<<(E + 255) / 256, 256, 0, stream>>>(edst, emax, ebuf, denom, E);
  // 5) normalized scatter-add aggregation (one wave per edge)
  {
    long long threads = (long long)E * 32;
    int blocks = (int)((threads + 255) / 256);
    gat_aggregate<<<blocks, 256, 0, stream>>>(feat_src, ebuf, denom, esrc, edst,
                                              out, E);
  }
}